// ExtractPatchesPositionLayer_42984032698838
// MI455X (gfx1250) — compile-verified
//
#include <hip/hip_runtime.h>
#include <stdint.h>

// Problem constants (from reference setup_inputs)
#define B_NPAD 522      // padded input H/W
#define B_NOUT 512      // output H/W
#define B_PAD  5        // halo pad in reference

// Tiling
#define TW 64           // output tile width
#define TH 32           // output tile height
#define LW (TW + 3)     // LDS tile width  (halo: -1 .. TW+1)
#define LH (TH + 3)     // LDS tile height (halo: -1 .. TH+1)
#define LDS_ELEMS (LW * LH)   // 2345 floats = 9380 B

#define AS1 __attribute__((address_space(1)))
#define AS3 __attribute__((address_space(3)))

#if defined(__HIP_DEVICE_COMPILE__)
#  if __has_builtin(__builtin_amdgcn_global_load_async_to_lds_b32)
#    define HAVE_ASYNC_LDS_BUILTIN 1
#  endif
#  if __has_builtin(__builtin_amdgcn_s_wait_asynccnt)
#    define HAVE_WAIT_ASYNC_BUILTIN 1
#  endif
#endif

// One 4-byte async copy global -> LDS (per-lane). CDNA5 ASYNCcnt path.
__device__ __forceinline__ void async_g2l_b32(float* lds_dst, const float* g_src) {
#if defined(__HIP_DEVICE_COMPILE__)
#  if defined(HAVE_ASYNC_LDS_BUILTIN)
    __builtin_amdgcn_global_load_async_to_lds_b32(
        (AS1 int*)(uintptr_t)g_src,
        (AS3 int*)(uint32_t)(uintptr_t)lds_dst,
        /*imm offset*/0, /*cpol*/0);
#  else
    // CDNA5 ISA; vdst = LDS byte offset VGPR, vaddr = 64-bit global addr
    asm volatile("global_load_async_to_lds_b32 %0, %1, off"
                 :
                 : "v"((uint32_t)(uintptr_t)lds_dst),
                   "v"((uint64_t)(uintptr_t)g_src)
                 : "memory");
#  endif
#endif
}

__device__ __forceinline__ void wait_async_zero() {
#if defined(__HIP_DEVICE_COMPILE__)
#  if defined(HAVE_WAIT_ASYNC_BUILTIN)
    __builtin_amdgcn_s_wait_asynccnt(0);
#  else
    asm volatile("s_wait_asynccnt 0x0" ::: "memory");
#  endif
#endif
}

__global__ __launch_bounds__(256)
void ExtractPatchesPositionLayer_42984032698838_kernel(const float* __restrict__ img,
                                                       const float* __restrict__ pos,
                                                       float* __restrict__ out)
{
    __shared__ float tile[LDS_ELEMS];

    const int b   = blockIdx.z;
    const int gx0 = blockIdx.x * TW;
    const int gy0 = blockIdx.y * TH;
    const int tid = threadIdx.x;

    // per-image sub-pixel offsets (uniform -> scalar loads)
    const float px = pos[2 * b + 0];
    const float py = pos[2 * b + 1];

    // block-level integer shift of the sampling window
    const int ox0 = (int)floorf(px + (float)B_PAD);
    const int oy0 = (int)floorf(py + (float)B_PAD);

    const int ibase = b * (B_NPAD * B_NPAD);

    // ---- stage (TH+3)x(TW+3) halo tile into LDS via async global->LDS ----
    // Addresses clamped into the image; border zeroing is applied later via
    // the bilinear weight masks, matching the reference's per-tap validity.
    for (int idx = tid; idx < LDS_ELEMS; idx += 256) {
        const int r = idx / LW;
        const int c = idx - r * LW;
        int gyL = gy0 + oy0 + r - 1;
        int gxL = gx0 + ox0 + c - 1;
        gyL = min(max(gyL, 0), B_NPAD - 1);
        gxL = min(max(gxL, 0), B_NPAD - 1);
        async_g2l_b32(&tile[idx], img + (ibase + gyL * B_NPAD + gxL));
    }
    wait_async_zero();
    __syncthreads();

    // ---- compute: 64 columns x 4 row-strips, 8 rows per thread ----
    const int tx  = tid & (TW - 1);
    const int tyq = tid >> 6;
    const int gx  = gx0 + tx;

    // x side: exact reference expression (float)(col) + px, col = gx + PAD
    const float xf  = (float)(gx + B_PAD) + px;
    const float x0f = floorf(xf);
    const float wx  = xf - x0f;
    const int   ix0 = (int)x0f;
    int c = ix0 - (gx0 + ox0) + 1;          // column in LDS tile (halo-adjusted)
    c = min(max(c, 0), LW - 2);
    const float mx0 = (ix0 >= 0 && ix0 < B_NPAD) ? 1.0f : 0.0f;
    const float mx1 = ((ix0 + 1) >= 0 && (ix0 + 1) < B_NPAD) ? 1.0f : 0.0f;
    const float cx0 = (1.0f - wx) * mx0;
    const float cx1 = wx * mx1;

    float* __restrict__ ocol = out + ((b * B_NOUT + gy0) * B_NOUT + gx);

#pragma unroll
    for (int k = 0; k < 8; ++k) {
        const int ty = tyq * 8 + k;
        const int gy = gy0 + ty;

        const float yf  = (float)(gy + B_PAD) + py;
        const float y0f = floorf(yf);
        const float wy  = yf - y0f;
        const int   iy0 = (int)y0f;
        int r = iy0 - (gy0 + oy0) + 1;      // row in LDS tile (halo-adjusted)
        r = min(max(r, 0), LH - 2);
        const float my0 = (iy0 >= 0 && iy0 < B_NPAD) ? 1.0f : 0.0f;
        const float my1 = ((iy0 + 1) >= 0 && (iy0 + 1) < B_NPAD) ? 1.0f : 0.0f;
        const float cy0 = (1.0f - wy) * my0;
        const float cy1 = wy * my1;

        const float v00 = tile[r * LW + c];
        const float v01 = tile[r * LW + c + 1];
        const float v10 = tile[(r + 1) * LW + c];
        const float v11 = tile[(r + 1) * LW + c + 1];

        const float top = v00 * cx0 + v01 * cx1;
        const float bot = v10 * cx0 + v11 * cx1;
        const float res = top * cy0 + bot * cy1;

        // write-once output: non-temporal store, keep L2 for input halo reuse
        __builtin_nontemporal_store(res, ocol + ty * B_NOUT);
    }
}

extern "C" void kernel_launch(void* const* d_in, const int* in_sizes, int n_in,
                              void* d_out, int out_size, void* d_ws, size_t ws_size,
                              hipStream_t stream) {
    const float* img = (const float*)d_in[0];   // (B, 522, 522, 1) fp32
    const float* pos = (const float*)d_in[1];   // (B, 2) fp32
    float* out = (float*)d_out;                 // (B, 512, 512, 1) fp32

    const int B = (n_in > 1 && in_sizes[1] > 0) ? (in_sizes[1] / 2) : 256;

    dim3 grid(B_NOUT / TW, B_NOUT / TH, B);     // (8, 16, B)
    dim3 block(256);
    hipLaunchKernelGGL(ExtractPatchesPositionLayer_42984032698838_kernel,
                       grid, block, 0, stream, img, pos, out);
}